// ScalarSoftmaxQuantization_55834574848320
// MI455X (gfx1250) — compile-verified
//
#include <hip/hip_runtime.h>

// ScalarSoftmaxQuantization, bandwidth-bound elementwise kernel for gfx1250.
// 88 MB single-touch traffic -> ~3.8 us floor at 23.3 TB/s. WMMA deliberately
// unused (K=4 einsum = 7 FMAs/element). Path: B128 NT loads/stores in two
// per-wave-contiguous streams, all memory ops in saddr+v0 scale_offset form
// (pre-offset base pointers, no per-thread 64-bit address math), SGPR-uniform
// bins, v_exp_f32 + v_rcp_f32 on the TRANS pipe (co-executes with VALU).

typedef float v4f __attribute__((ext_vector_type(4)));

// ALPHA * log2(e): exp(ALPHA*d) == exp2(ALPHA_LOG2E * d)
#define ALPHA_LOG2E (-50.0f * 1.44269504088896340736f)

__device__ __forceinline__ float soft_quant(float x, float b0, float b1,
                                            float b2, float b3) {
    float d0 = __builtin_fabsf(x - b0);
    float d1 = __builtin_fabsf(x - b1);
    float d2 = __builtin_fabsf(x - b2);
    float d3 = __builtin_fabsf(x - b3);
    // softmax stabilization: max(ALPHA*d) == ALPHA*min(d) since ALPHA < 0.
    // Keeping the explicit (d - m) form makes the dominant weight exactly 1.0
    // (d_min - m == 0), matching the reference's stabilized softmax.
    float m  = fminf(fminf(d0, d1), fminf(d2, d3));
    // v_exp_f32 (TRANS pipe, co-executes with VALU)
    float w0 = __builtin_amdgcn_exp2f(ALPHA_LOG2E * (d0 - m));
    float w1 = __builtin_amdgcn_exp2f(ALPHA_LOG2E * (d1 - m));
    float w2 = __builtin_amdgcn_exp2f(ALPHA_LOG2E * (d2 - m));
    float w3 = __builtin_amdgcn_exp2f(ALPHA_LOG2E * (d3 - m));
    float num = __builtin_fmaf(w0, b0,
                __builtin_fmaf(w1, b1,
                __builtin_fmaf(w2, b2, w3 * b3)));
    float den = (w0 + w1) + (w2 + w3);   // in [1,4], never zero
    // v_rcp_f32 + one Newton-Raphson step: ~1 ulp, avoids the IEEE-div
    // v_div_scale/v_div_fmas/v_div_fixup VALU sequence.
    float r = __builtin_amdgcn_rcpf(den);
    r = __builtin_fmaf(__builtin_fmaf(-den, r, 1.0f), r, r);
    return num * r;
}

__device__ __forceinline__ v4f soft_quant4(v4f v, float b0, float b1,
                                           float b2, float b3) {
    v4f r;
    r.x = soft_quant(v.x, b0, b1, b2, b3);
    r.y = soft_quant(v.y, b0, b1, b2, b3);
    r.z = soft_quant(v.z, b0, b1, b2, b3);
    r.w = soft_quant(v.w, b0, b1, b2, b3);
    return r;
}

__global__ __launch_bounds__(256) void ssq_kernel(
    const v4f* __restrict__ xa, const v4f* __restrict__ xb,
    v4f* __restrict__ oa, v4f* __restrict__ ob,
    const float* __restrict__ bins, unsigned npair,
    const float* __restrict__ xt, float* __restrict__ ot, unsigned ntail) {
    // Uniform bin centers -> s_load_b128 (scalar path)
    const float b0 = bins[0];
    const float b1 = bins[1];
    const float b2 = bins[2];
    const float b3 = bins[3];

    const unsigned i = blockIdx.x * blockDim.x + threadIdx.x;

    if (i < npair) {
        // Two independent per-wave-contiguous B128 NT streams; pre-offset
        // SGPR bases mean every access is saddr + v0 scale_offset (no 64-bit
        // VALU address math), and the two in-flight loads hide latency.
        v4f va = __builtin_nontemporal_load(xa + i);
        v4f vb = __builtin_nontemporal_load(xb + i);
        v4f ra = soft_quant4(va, b0, b1, b2, b3);
        v4f rb = soft_quant4(vb, b0, b1, b2, b3);
        __builtin_nontemporal_store(ra, oa + i);
        __builtin_nontemporal_store(rb, ob + i);
    }

    // Scalar tail, ntail < 8 (never taken for the reference shape: n % 8 == 0)
    if (i < ntail) {
        ot[i] = soft_quant(xt[i], b0, b1, b2, b3);
    }
}

extern "C" void kernel_launch(void* const* d_in, const int* in_sizes, int n_in,
                              void* d_out, int out_size, void* d_ws, size_t ws_size,
                              hipStream_t stream) {
    const float* x    = (const float*)d_in[0];   // [2048, 21, 256] f32
    const float* bins = (const float*)d_in[1];   // [4] f32
    float* out        = (float*)d_out;           // [2048, 21, 256] f32

    const unsigned n     = (unsigned)out_size;  // 11,010,048
    const unsigned npair = n >> 3;              // 1,376,256 float4-pairs
    const unsigned ntail = n & 7u;              // 0 for the reference shape

    // Stream A: float4s [0, npair); stream B: float4s [npair, 2*npair)
    const v4f* xa = (const v4f*)x;
    const v4f* xb = xa + npair;
    v4f* oa       = (v4f*)out;
    v4f* ob       = oa + npair;
    // Tail bases (elements [8*npair, n))
    const float* xt = x + (size_t)npair * 8u;
    float* ot       = out + (size_t)npair * 8u;

    const unsigned threads = 256;               // 8 wave32s per block
    unsigned work   = npair > ntail ? npair : ntail;
    unsigned blocks = (work + threads - 1) / threads;  // 5376 for ref shape
    if (blocks < 1) blocks = 1;

    hipLaunchKernelGGL(ssq_kernel, dim3(blocks), dim3(threads), 0, stream,
                       xa, xb, oa, ob, bins, npair, xt, ot, ntail);
}